// GcnNet_64991445123451
// MI455X (gfx1250) — compile-verified
//
#include <hip/hip_runtime.h>
#include <hip/hip_bf16.h>

typedef __attribute__((ext_vector_type(2))) float v2f;
typedef __attribute__((ext_vector_type(8))) float v8f;

#define HID 64
#define NG  128

__device__ __forceinline__ void atomAdd(float* p, float v) {
    __hip_atomic_fetch_add(p, v, __ATOMIC_RELAXED, __HIP_MEMORY_SCOPE_AGENT);
}

// ---------------- utility fills ----------------
__global__ void k_fill(float* __restrict__ p, float v, int n) {
    int i = blockIdx.x * blockDim.x + threadIdx.x;
    if (i < n) p[i] = v;
}

// ---------------- degree / norm ----------------
__global__ void k_degree(const int* __restrict__ ei, float* __restrict__ deg, int E) {
    int e = blockIdx.x * blockDim.x + threadIdx.x;
    if (e < E) atomAdd(&deg[ei[E + e]], 1.0f);   // dst row of edge_index
}

__global__ void k_rsqrt_inplace(float* __restrict__ d, int n) {
    int i = blockIdx.x * blockDim.x + threadIdx.x;
    if (i < n) { float v = d[i]; d[i] = (v > 0.0f) ? __frsqrt_rn(v) : 0.0f; }
}

// ---------------- fp32 WMMA GEMM: Out[n x 64] = A[n x 64] @ W[64 x 64] ----------------
// One wave per 16x16 output tile; 4 waves/block cover all 64 output columns of one M tile.
__global__ __launch_bounds__(128) void k_gemm_wmma(const float* __restrict__ A,
                                                   const float* __restrict__ W,
                                                   float* __restrict__ Out,
                                                   int mtiles) {
    int mt = blockIdx.x;
    if (mt >= mtiles) return;
    int wave = threadIdx.x >> 5;        // 0..3 -> N tile
    int lane = threadIdx.x & 31;
    int l15  = lane & 15;
    int hi   = lane >> 4;               // 0 or 1 (selects K pair per ISA layout)
    int m_base = mt * 16;
    int n_base = wave * 16;

    const float* __restrict__ Arow = A + (size_t)(m_base + l15) * HID;
    v8f c = {};
#pragma unroll
    for (int k0 = 0; k0 < HID; k0 += 4) {
        int kk = k0 + (hi << 1);
        v2f a, b;
        // A 16x4: lanes 0-15 -> K=k0,k0+1 ; lanes 16-31 -> K=k0+2,k0+3 (row = l15)
        a.x = Arow[kk];
        a.y = Arow[kk + 1];
        // B 4x16: row K striped across lanes within a VGPR (col = n_base + l15)
        b.x = W[(size_t)kk * HID + n_base + l15];
        b.y = W[(size_t)(kk + 1) * HID + n_base + l15];
        c = __builtin_amdgcn_wmma_f32_16x16x4_f32(false, a, false, b, (short)0, c,
                                                  false, false);
    }
    // C/D layout: VGPR v -> row m_base + hi*8 + v, col n_base + l15
    int n = n_base + l15;
#pragma unroll
    for (int v = 0; v < 8; ++v)
        Out[(size_t)(m_base + hi * 8 + v) * HID + n] = c[v];
}

// Scalar tail for rows not covered by full 16-row tiles (not used when n % 16 == 0).
__global__ void k_gemm_tail(const float* __restrict__ A, const float* __restrict__ W,
                            float* __restrict__ Out, int row0, int n) {
    int idx = blockIdx.x * blockDim.x + threadIdx.x;
    int rows = n - row0;
    if (idx >= rows * HID) return;
    int r = row0 + idx / HID, cidx = idx % HID;
    float acc = 0.0f;
    for (int k = 0; k < HID; ++k) acc += A[(size_t)r * HID + k] * W[(size_t)k * HID + cidx];
    Out[(size_t)r * HID + cidx] = acc;
}

// ---------------- self-loop term (also initializes agg) ----------------
__global__ void k_selfloop(const float* __restrict__ hW, const float* __restrict__ dinv,
                           float* __restrict__ agg, int total) {
    int idx = blockIdx.x * blockDim.x + threadIdx.x;
    if (idx >= total) return;
    int i = idx >> 6;
    float di = dinv[i];
    agg[idx] = hW[idx] * di * di;
}

// ---------------- edge scatter: 16 lanes per edge, float4 per lane ----------------
__global__ __launch_bounds__(256) void k_scatter(const float* __restrict__ hW,
                                                 const int* __restrict__ ei,
                                                 const float* __restrict__ dinv,
                                                 float* __restrict__ agg, int E) {
    int t = blockIdx.x * 256 + threadIdx.x;
    int e = t >> 4;
    if (e >= E) return;
    int j = (t & 15) << 2;                 // feature base: 0,4,...,60
    int s = ei[e];                         // src
    int d = ei[E + e];                     // dst
    float nrm = dinv[s] * dinv[d];
    const float4 v = *reinterpret_cast<const float4*>(hW + (size_t)s * HID + j);
    float* o = agg + (size_t)d * HID + j;
    atomAdd(o + 0, v.x * nrm);
    atomAdd(o + 1, v.y * nrm);
    atomAdd(o + 2, v.z * nrm);
    atomAdd(o + 3, v.w * nrm);
}

// ---------------- bias + relu ----------------
__global__ void k_bias_relu(const float* __restrict__ agg, const float* __restrict__ b,
                            float* __restrict__ out, int total) {
    int idx = blockIdx.x * blockDim.x + threadIdx.x;
    if (idx >= total) return;
    out[idx] = fmaxf(agg[idx] + b[idx & 63], 0.0f);
}

// ---------------- mean pool (atomic accumulate) ----------------
__global__ void k_pool(const float* __restrict__ h, const int* __restrict__ batch,
                       float* __restrict__ sums, float* __restrict__ counts, int n) {
    int idx = blockIdx.x * blockDim.x + threadIdx.x;
    if (idx >= n * HID) return;
    int i = idx >> 6, f = idx & 63;
    int g = batch[i];
    atomAdd(&sums[(size_t)g * HID + f], h[idx]);
    if (f == 0) atomAdd(&counts[g], 1.0f);
}

// ---------------- MLP head: one thread per graph ----------------
__global__ __launch_bounds__(NG) void k_head(const float* __restrict__ sums,
                                             const float* __restrict__ counts,
                                             const float* __restrict__ fc1w,
                                             const float* __restrict__ fc1b,
                                             const float* __restrict__ fc2w,
                                             const float* __restrict__ fc2b,
                                             float* __restrict__ out, int ngraphs) {
    int g = blockIdx.x * blockDim.x + threadIdx.x;
    if (g >= ngraphs) return;
    float inv = 1.0f / fmaxf(counts[g], 1.0f);
    float p[HID];
#pragma unroll
    for (int k = 0; k < HID; ++k) p[k] = sums[(size_t)g * HID + k] * inv;
    float acc = fc2b[0];
#pragma unroll
    for (int j = 0; j < 10; ++j) {
        float t = fc1b[j];
#pragma unroll
        for (int k = 0; k < HID; ++k) t += p[k] * fc1w[k * 10 + j];
        acc += fmaxf(t, 0.0f) * fc2w[j];
    }
    out[g] = acc;
}

// =========================================================================
extern "C" void kernel_launch(void* const* d_in, const int* in_sizes, int n_in,
                              void* d_out, int out_size, void* d_ws, size_t ws_size,
                              hipStream_t stream) {
    const float* x     = (const float*)d_in[0];
    const int*   ei    = (const int*)d_in[1];   // [2, E] (int per harness contract)
    const int*   batch = (const int*)d_in[2];
    const float* W1 = (const float*)d_in[3];  const float* b1 = (const float*)d_in[4];
    const float* W2 = (const float*)d_in[5];  const float* b2 = (const float*)d_in[6];
    const float* W3 = (const float*)d_in[7];  const float* b3 = (const float*)d_in[8];
    const float* fc1w = (const float*)d_in[9];  const float* fc1b = (const float*)d_in[10];
    const float* fc2w = (const float*)d_in[11]; const float* fc2b = (const float*)d_in[12];
    float* out = (float*)d_out;

    const int n  = in_sizes[0] / HID;       // 100000
    const int E  = in_sizes[1] / 2;         // 1600000
    const int ng = out_size;                // 128
    const int nh = n * HID;

    // Workspace layout (floats, 64-float aligned slots)
    size_t off = 0;
    auto alloc = [&](size_t cnt) {
        float* p = (float*)d_ws + off;
        off += (cnt + 63) & ~(size_t)63;
        return p;
    };
    float* deg   = alloc(n);        // degree -> dinv (in place)
    float* H0    = alloc(nh);       // GEMM output (h @ W)
    float* H1    = alloc(nh);       // aggregation buffer
    float* H2    = alloc(nh);       // activated hidden state
    float* sums  = alloc((size_t)ng * HID);
    float* cnts  = alloc(ng);

    const int B = 256;
    auto g1 = [&](int total) { return dim3((total + B - 1) / B); };

    // degrees with self loop: deg = 1 + sum over dst
    k_fill<<<g1(n), B, 0, stream>>>(deg, 1.0f, n);
    k_degree<<<g1(E), B, 0, stream>>>(ei, deg, E);
    k_rsqrt_inplace<<<g1(n), B, 0, stream>>>(deg, n);

    const int mtiles = n / 16;
    const int mrem0  = mtiles * 16;

    const float* hin = x;
    const float* Ws[3] = {W1, W2, W3};
    const float* bs[3] = {b1, b2, b3};
    for (int l = 0; l < 3; ++l) {
        k_gemm_wmma<<<dim3(mtiles), dim3(128), 0, stream>>>(hin, Ws[l], H0, mtiles);
        if (mrem0 < n) {
            int rem = (n - mrem0) * HID;
            k_gemm_tail<<<g1(rem), B, 0, stream>>>(hin, Ws[l], H0, mrem0, n);
        }
        k_selfloop<<<g1(nh), B, 0, stream>>>(H0, deg, H1, nh);
        k_scatter<<<g1(E * 16), B, 0, stream>>>(H0, ei, deg, H1, E);
        k_bias_relu<<<g1(nh), B, 0, stream>>>(H1, bs[l], H2, nh);
        hin = H2;
    }

    k_fill<<<g1(ng * HID), B, 0, stream>>>(sums, 0.0f, ng * HID);
    k_fill<<<g1(ng), B, 0, stream>>>(cnts, 0.0f, ng);
    k_pool<<<g1(nh), B, 0, stream>>>(H2, batch, sums, cnts, n);
    k_head<<<g1(ng), NG, 0, stream>>>(sums, cnts, fc1w, fc1b, fc2w, fc2b, out, ng);
}